// TopDownHTMM_15564961481297
// MI455X (gfx1250) — compile-verified
//
#include <hip/hip_runtime.h>

typedef __attribute__((ext_vector_type(2))) float v2f;
typedef __attribute__((ext_vector_type(8))) float v8f;

#define NTREE  512
#define DEPTH  10
#define NSTATE 8
#define NSYM   200
#define NGEN   8
#define NPT    1023

// ---- workspace (float) layout ----
#define WSA     0        // 512 : softmax A  [(j*8+i)*8+g]
#define WSPI    512      // 64  : softmax Pi [c*8+g]
#define WSPRIOR 576      // 640 : prior per level [(l*8+c)*8+g]
#define WSB     1216     // 12800 : softmax B [(c*200+m)*8+g]

// ---- dynamic LDS (float) layout ----
#define OFF_BETA_A 0       // 512*64
#define OFF_BETA_B 32768   // 256*64
#define OFF_UV     49152   // 8 waves * 16*16
#define OFF_AN     51200   // 512
#define OFF_PRIOR  51712   // 640
#define OFF_LL     52352   // 8
#define SMEM_FLOATS 52360  // ~209 KB < 320 KB/WGP

// ds_swizzle group-of-32 XOR butterfly: offset = and_mask(0x1F) | xor_mask<<10
template <int PAT>
__device__ __forceinline__ float xor_add(float v)
{
    const int sw = __builtin_amdgcn_ds_swizzle(__builtin_bit_cast(int, v), PAT);
    return v + __builtin_bit_cast(float, sw);
}

__global__ void htmm_prep(const float* __restrict__ lA,
                          const float* __restrict__ lB,
                          const float* __restrict__ lPi,
                          float* __restrict__ ws)
{
    const int t = threadIdx.x;
    if (t < 64) {                       // A: softmax over j for each (i,g)
        const int i = t >> 3, g = t & 7;
        float mx = -1e30f;
        for (int j = 0; j < 8; ++j) mx = fmaxf(mx, lA[(j*8+i)*8+g]);
        float s = 0.f;
        for (int j = 0; j < 8; ++j) { float e = __expf(lA[(j*8+i)*8+g]-mx); ws[WSA+(j*8+i)*8+g]=e; s+=e; }
        const float inv = 1.f/s;
        for (int j = 0; j < 8; ++j) ws[WSA+(j*8+i)*8+g] *= inv;
    }
    if (t < 8) {                        // Pi: softmax over c per g
        const int g = t;
        float mx = -1e30f;
        for (int c = 0; c < 8; ++c) mx = fmaxf(mx, lPi[c*8+g]);
        float s = 0.f;
        for (int c = 0; c < 8; ++c) { float e = __expf(lPi[c*8+g]-mx); ws[WSPI+c*8+g]=e; s+=e; }
        const float inv = 1.f/s;
        for (int c = 0; c < 8; ++c) ws[WSPI+c*8+g] *= inv;
    }
    if (t < 64) {                       // B: softmax over m per (c,g)
        const int c = t >> 3, g = t & 7;
        float mx = -1e30f;
        for (int m = 0; m < NSYM; ++m) mx = fmaxf(mx, lB[(c*NSYM+m)*8+g]);
        float s = 0.f;
        for (int m = 0; m < NSYM; ++m) { float e = __expf(lB[(c*NSYM+m)*8+g]-mx); ws[WSB+(c*NSYM+m)*8+g]=e; s+=e; }
        const float inv = 1.f/s;
        for (int m = 0; m < NSYM; ++m) ws[WSB+(c*NSYM+m)*8+g] *= inv;
    }
    __syncthreads();
    const int j = t >> 3, g = t & 7;    // prior_lvl[l] = A^l * Pi
    if (t < 64) ws[WSPRIOR + (0*8+j)*8+g] = ws[WSPI + j*8+g];
    for (int l = 1; l < DEPTH; ++l) {
        __syncthreads();
        float v = 0.f;
        if (t < 64)
            for (int i = 0; i < 8; ++i)
                v += ws[WSA+(j*8+i)*8+g] * ws[WSPRIOR+((l-1)*8+i)*8+g];
        __syncthreads();
        if (t < 64) ws[WSPRIOR+(l*8+j)*8+g] = v;
    }
}

__launch_bounds__(256, 1)
__global__ void htmm_tree(const int* __restrict__ x,
                          const float* __restrict__ ws,
                          float* __restrict__ out)
{
    extern __shared__ float sm[];
    const int tid  = threadIdx.x;
    const int lane = tid & 31;
    const int wave = tid >> 5;
    const int tree = blockIdx.x;
    const int xbase = tree * NPT;

    float* betaA = sm + OFF_BETA_A;
    float* betaB = sm + OFF_BETA_B;
    float* uvb   = sm + OFF_UV + wave * 256;
    float* An    = sm + OFF_AN;
    float* pri   = sm + OFF_PRIOR;
    float* ll    = sm + OFF_LL;
    const float* Bn = ws + WSB;

    for (int i = tid; i < 512; i += 256) An[i]  = ws[WSA + i];
    for (int i = tid; i < 640; i += 256) pri[i] = ws[WSPRIOR + i];
    if (tid < NGEN) ll[tid] = 0.f;
    __syncthreads();

    // ---- leaves (depth 9): beta = prior*emit, normalize, accumulate log(nu) ----
    {
        const int g = tid & 7;          // fixed per thread (stride 256 % 8 == 0)
        float acc = 0.f;
        for (int task = tid; task < 512 * 8; task += 256) {
            const int leaf = task >> 3;
            const int xv = x[xbase + 511 + leaf];
            float v[8]; float nu = 0.f;
#pragma unroll
            for (int i = 0; i < 8; ++i) {
                v[i] = pri[(9*8 + i)*8 + g] * Bn[(i*NSYM + xv)*8 + g];
                nu += v[i];
            }
            const float inv = 1.f / nu;
#pragma unroll
            for (int i = 0; i < 8; ++i) betaA[(g*8 + i)*512 + leaf] = v[i] * inv;
            acc += __logf(nu);
        }
        atomicAdd(&ll[g], acc);
    }
    __syncthreads();

    // ---- upward: one wave per gen, level ping-pong in LDS ----
    float* cbuf = betaA;
    float* pbuf = betaB;
    const int g    = wave;
    const int m    = lane & 15;
    const int half = lane >> 4;
    const int jA   = half * 2;          // K-pair owned by this lane half

    // B operand = normalized A[j,i,g]; invariant across levels; cols >= 8 zeroed
    v2f b1, b2;
    b1.x = (m < 8) ? An[(jA+0)*64 + m*8 + g] : 0.f;
    b1.y = (m < 8) ? An[(jA+1)*64 + m*8 + g] : 0.f;
    b2.x = (m < 8) ? An[(jA+4)*64 + m*8 + g] : 0.f;
    b2.y = (m < 8) ? An[(jA+5)*64 + m*8 + g] : 0.f;

    for (int k = 8; k >= 0; --k) {
        const int P  = 1 << k;
        const int Ch = P << 1;
        const int nchunk = (Ch + 15) >> 4;
        const int rows = (Ch < 16) ? Ch : 16;

        float rp[8];
#pragma unroll
        for (int j = 0; j < 8; ++j) rp[j] = 1.f / pri[((k+1)*8 + j)*8 + g];

        for (int chunk = 0; chunk < nchunk; ++chunk) {
            const int cb   = chunk << 4;
            const int cm   = cb + m;
            // clamp index (always a safe LDS address), zero padded rows via mask
            const float msk = (cm < Ch) ? 1.f : 0.f;
            const int   ci  = (cm < Ch) ? cm : 0;

            // A operand = ratio rows (16 sibling children x K=8, split K=4+4)
            v2f a1, a2;
            a1.x = cbuf[(g*8 + jA+0)*Ch + ci] * (rp[jA+0] * msk);
            a1.y = cbuf[(g*8 + jA+1)*Ch + ci] * (rp[jA+1] * msk);
            a2.x = cbuf[(g*8 + jA+4)*Ch + ci] * (rp[jA+4] * msk);
            a2.y = cbuf[(g*8 + jA+5)*Ch + ci] * (rp[jA+5] * msk);

            v8f d = {0.f,0.f,0.f,0.f,0.f,0.f,0.f,0.f};
            d = __builtin_amdgcn_wmma_f32_16x16x4_f32(false, a1, false, b1, (short)0, d, false, false);
            d = __builtin_amdgcn_wmma_f32_16x16x4_f32(false, a2, false, b2, (short)0, d, false, false);

#pragma unroll
            for (int r = 0; r < 8; ++r)
                uvb[(r + half*8)*16 + m] = d[r];
            asm volatile("s_wait_dscnt 0" ::: "memory");   // wave-private LDS RAW

            // combine pairs of children into 8 parents (2 passes x 32 lanes)
#pragma unroll
            for (int pass = 0; pass < 2; ++pass) {
                const int idx = lane + pass * 32;
                const int pl  = idx >> 3;
                const int i   = idx & 7;
                const bool valid = (2*pl + 1) < rows;
                const int p = (cb >> 1) + pl;
                const float uv0 = uvb[(2*pl)*16 + i];
                const float uv1 = uvb[(2*pl+1)*16 + i];
                float unn = 0.f;
                if (valid) {
                    const int xv = x[xbase + (P - 1) + p];
                    const float b0 = pri[(k*8 + i)*8 + g] * Bn[(i*NSYM + xv)*8 + g];
                    unn = b0 * b0 * uv0 * uv1;   // reference includes prev_beta twice
                }
                // reduce over i (low 3 lane bits) with XOR swizzle butterflies
                float nu = unn;
                nu = xor_add<0x041F>(nu);   // xor 1
                nu = xor_add<0x081F>(nu);   // xor 2
                nu = xor_add<0x101F>(nu);   // xor 4
                if (valid) {
                    pbuf[(g*8 + i)*P + p] = unn / nu;
                    if (i == 0) atomicAdd(&ll[g], __logf(nu));
                }
            }
            asm volatile("s_wait_dscnt 0" ::: "memory");
        }
        float* t = cbuf; cbuf = pbuf; pbuf = t;
    }

    __syncthreads();
    if (tid < NGEN) out[tree * NGEN + tid] = ll[tid];
}

extern "C" void kernel_launch(void* const* d_in, const int* in_sizes, int n_in,
                              void* d_out, int out_size, void* d_ws, size_t ws_size,
                              hipStream_t stream)
{
    (void)in_sizes; (void)out_size; (void)ws_size;
    // setup_inputs order; index lambdas from the end (lists may flatten)
    const int*   x   = (const int*)d_in[0];
    const float* lA  = (const float*)d_in[n_in - 3];
    const float* lB  = (const float*)d_in[n_in - 2];
    const float* lPi = (const float*)d_in[n_in - 1];
    float* ws  = (float*)d_ws;
    float* out = (float*)d_out;

    const size_t smem = SMEM_FLOATS * sizeof(float);
    (void)hipFuncSetAttribute((const void*)htmm_tree,
                              hipFuncAttributeMaxDynamicSharedMemorySize, (int)smem);

    htmm_prep<<<1, 256, 0, stream>>>(lA, lB, lPi, ws);
    htmm_tree<<<NTREE, 256, smem, stream>>>(x, ws, out);
}